// SSIMLoss_26456998543621
// MI455X (gfx1250) — compile-verified
//
#include <hip/hip_runtime.h>
#include <cstdint>

// ---------------------------------------------------------------------------
// SSIM loss for MI455X (gfx1250, wave32).
// Pipeline:
//   k0: init accumulators in ws
//   k1: amplitude |Xt*s+m|, |Yt*s+m| -> ws planes (192,256,256) + global max
//   k2: per 16x16 output tile: 7x7 box stats via V_WMMA_F32_16X16X4_F32
//       (stencil-as-banded-GEMM), SSIM map, wave reduce, f64 atomic add
//   k3: out = 1 - sum / (192*250*250)
// ---------------------------------------------------------------------------

typedef __attribute__((ext_vector_type(2))) float v2f;
typedef __attribute__((ext_vector_type(8))) float v8f;

#define BB    16
#define HH    256
#define WW    256
#define TT    12
#define NIMG  (BB * TT)            // 192
#define HW    (HH * WW)            // 65536
#define AMPN  (NIMG * HW)          // 12,582,912 per field
#define WIN   7
#define ODIM  (HH - WIN + 1)       // 250
#define APR   (16 + WIN - 1)       // 22 input rows/cols per tile

// ---------------- k0: init ----------------
__global__ void ssim_init_kernel(unsigned* __restrict__ maxbits,
                                 double* __restrict__ accum) {
  if (threadIdx.x == 0) {
    maxbits[0] = 0u;
    maxbits[1] = 0u;
    *accum = 0.0;
  }
}

// ---------------- k1: amplitude + global max ----------------
__global__ __launch_bounds__(256)
void ssim_amp_kernel(const float* __restrict__ Xt, const float* __restrict__ Yt,
                     const float* __restrict__ mean, const float* __restrict__ stdv,
                     float* __restrict__ Xa, float* __restrict__ Ya,
                     unsigned* __restrict__ maxbits) {
  int i = blockIdx.x * blockDim.x + threadIdx.x;
  float ax = 0.f, ay = 0.f;
  if (i < AMPN) {
    int b    = i / (HW * TT);
    int rem  = i - b * (HW * TT);
    int h    = rem / (WW * TT);
    int rem2 = rem - h * (WW * TT);
    int w    = rem2 / TT;
    int t    = rem2 - w * TT;
    float m = mean[b], s = stdv[b];
    size_t ii = (size_t)i * 2;                       // (b,h,w,t,2) flat
    float x0 = fmaf(Xt[ii], s, m), x1 = fmaf(Xt[ii + 1], s, m);
    float y0 = fmaf(Yt[ii], s, m), y1 = fmaf(Yt[ii + 1], s, m);
    ax = sqrtf(x0 * x0 + x1 * x1);
    ay = sqrtf(y0 * y0 + y1 * y1);
    size_t o = ((size_t)(b * TT + t) * HH + h) * WW + w;   // (n,h,w) planes
    Xa[o] = ax;
    Ya[o] = ay;
  }
  // wave32 max reduce, one atomic per wave (amp >= 0 -> uint-monotonic bits)
  for (int off = 16; off > 0; off >>= 1) {
    ax = fmaxf(ax, __shfl_xor(ax, off, 32));
    ay = fmaxf(ay, __shfl_xor(ay, off, 32));
  }
  if ((threadIdx.x & 31) == 0) {
    atomicMax(&maxbits[0], __float_as_uint(ax));
    atomicMax(&maxbits[1], __float_as_uint(ay));
  }
}

// ---------------- k2: WMMA box filter + SSIM ----------------
// One wave per 16x16 output tile. Separable 7x7 sum as banded GEMM:
//   V(16x22) = Aband(16x22) * P(22x22)   [2 col groups x 6 K-steps of K=4]
//   D(16x16) = V(16x22)     * Bband(22x16) [6 K-steps]
// f32 WMMA layouts (ISA 7.12.2): A 16x4: lanes0-15 K={0,1}, lanes16-31 K={2,3};
// B 4x16 mirrored; D VGPR r: lanes0-15 M=r, lanes16-31 M=r+8, N=lane&15.
__global__ __launch_bounds__(32)
void ssim_tile_kernel(const float* __restrict__ Xa, const float* __restrict__ Ya,
                      const unsigned* __restrict__ maxbits,
                      double* __restrict__ accum) {
  __shared__ float sX[APR][32];
  __shared__ float sY[APR][32];
  __shared__ float sV[5][16][32];

  const int lane = threadIdx.x;          // 0..31, wave32
  const int bx   = blockIdx.x;
  const int n    = bx >> 8;              // image 0..191
  const int tid  = bx & 255;             // 16x16 tiles per image
  const int th   = (tid >> 4) * 16;      // output row base
  const int tw   = (tid & 15) * 16;      // output col base

  const float invX = 1.0f / __uint_as_float(maxbits[0]);
  const float invY = 1.0f / __uint_as_float(maxbits[1]);
  const float* Xn = Xa + (size_t)n * HW;
  const float* Yn = Ya + (size_t)n * HW;

  // Load 22x22 apron, columns padded to 32 with zeros; edges clamped (the
  // affected outputs are masked out of the reduction below).
  for (int r = 0; r < APR; ++r) {
    int gh = th + r;    if (gh > HH - 1) gh = HH - 1;
    int gw = tw + lane; if (gw > WW - 1) gw = WW - 1;
    float vx = 0.f, vy = 0.f;
    if (lane < APR) {
      vx = Xn[gh * WW + gw] * invX;
      vy = Yn[gh * WW + gw] * invY;
    }
    sX[r][lane] = vx;
    sY[r][lane] = vy;
  }
  __syncthreads();

  const int khalf = (lane >= 16) ? 2 : 0;  // K split across lane halves
  const int mrow  = lane & 15;             // row (A/D) or col (B) index

  // ---- vertical pass: for each stat, 2 column groups x 6 K-steps ----
  for (int s = 0; s < 5; ++s) {
    for (int g = 0; g < 2; ++g) {
      v8f acc = {};
      const int ncol = mrow + g * 16;
      for (int kk = 0; kk < 24; kk += 4) {     // K = 22 padded to 24
        const int k0 = kk + khalf, k1 = k0 + 1;
        v2f a, b;
        // band: Aband[m,k] = (m <= k < m+7); auto-zero for k >= 22
        a.x = (k0 >= mrow && k0 < mrow + WIN) ? 1.f : 0.f;
        a.y = (k1 >= mrow && k1 < mrow + WIN) ? 1.f : 0.f;
        float x0 = 0.f, y0 = 0.f, x1 = 0.f, y1 = 0.f;
        if (k0 < APR) { x0 = sX[k0][ncol]; y0 = sY[k0][ncol]; }
        if (k1 < APR) { x1 = sX[k1][ncol]; y1 = sY[k1][ncol]; }
        switch (s) {
          case 0:  b.x = x0;      b.y = x1;      break;  // X
          case 1:  b.x = y0;      b.y = y1;      break;  // Y
          case 2:  b.x = x0 * x0; b.y = x1 * x1; break;  // XX
          case 3:  b.x = y0 * y0; b.y = y1 * y1; break;  // YY
          default: b.x = x0 * y0; b.y = x1 * y1; break;  // XY
        }
        acc = __builtin_amdgcn_wmma_f32_16x16x4_f32(
                  false, a, false, b, (short)0, acc, false, false);
      }
      // corner turn: D -> LDS (cols >= 22 are exact zeros: zero-padded input)
      const int mbase = (lane < 16) ? 0 : 8;
      for (int r = 0; r < 8; ++r)
        sV[s][mbase + r][ncol] = acc[r];
    }
  }
  __syncthreads();

  // ---- horizontal pass: D_s = V_s(16x22) * Bband(22x16) ----
  v8f D[5];
  for (int s = 0; s < 5; ++s) {
    v8f acc = {};
    for (int kk = 0; kk < 24; kk += 4) {
      const int k0 = kk + khalf, k1 = k0 + 1;
      v2f a, b;
      a.x = sV[s][mrow][k0];                 // k <= 23 < 32, pad cols are 0
      a.y = sV[s][mrow][k1];
      b.x = (k0 >= mrow && k0 < mrow + WIN) ? 1.f : 0.f;
      b.y = (k1 >= mrow && k1 < mrow + WIN) ? 1.f : 0.f;
      acc = __builtin_amdgcn_wmma_f32_16x16x4_f32(
                false, a, false, b, (short)0, acc, false, false);
    }
    D[s] = acc;
  }

  // ---- SSIM map + masked reduction ----
  const float inv_np   = 1.0f / (WIN * WIN);                // box mean
  const float cov_norm = (float)(WIN * WIN) / (WIN * WIN - 1);
  const float C1 = 0.01f * 0.01f, C2 = 0.03f * 0.03f;
  const int   mbase = (lane < 16) ? 0 : 8;
  float part = 0.f;
  for (int r = 0; r < 8; ++r) {
    const int ho = th + mbase + r;
    const int wo = tw + mrow;
    if (ho < ODIM && wo < ODIM) {
      float ux  = D[0][r] * inv_np, uy  = D[1][r] * inv_np;
      float uxx = D[2][r] * inv_np, uyy = D[3][r] * inv_np;
      float uxy = D[4][r] * inv_np;
      float vx  = cov_norm * (uxx - ux * ux);
      float vy  = cov_norm * (uyy - uy * uy);
      float vxy = cov_norm * (uxy - ux * uy);
      float A1 = 2.f * ux * uy + C1, A2 = 2.f * vxy + C2;
      float B1 = ux * ux + uy * uy + C1, B2 = vx + vy + C2;
      part += (A1 * A2) / (B1 * B2);
    }
  }
  for (int off = 16; off > 0; off >>= 1)
    part += __shfl_xor(part, off, 32);
  if (lane == 0)
    atomicAdd(accum, (double)part);
}

// ---------------- k3: finalize ----------------
__global__ void ssim_final_kernel(const double* __restrict__ accum,
                                  float* __restrict__ out) {
  const double npix = (double)NIMG * (double)ODIM * (double)ODIM;
  out[0] = 1.0f - (float)(*accum / npix);
}

// ---------------------------------------------------------------------------
extern "C" void kernel_launch(void* const* d_in, const int* in_sizes, int n_in,
                              void* d_out, int out_size, void* d_ws, size_t ws_size,
                              hipStream_t stream) {
  (void)in_sizes; (void)n_in; (void)out_size; (void)ws_size;
  const float* Xt   = (const float*)d_in[0];
  const float* Yt   = (const float*)d_in[1];
  const float* mean = (const float*)d_in[2];
  const float* stdv = (const float*)d_in[3];
  float* out = (float*)d_out;

  // Workspace: Xa (48MB) | Ya (48MB) | accum f64 | max bits x2
  float*    Xa      = (float*)d_ws;
  float*    Ya      = Xa + AMPN;
  double*   accum   = (double*)(Ya + AMPN);        // byte offset 100663296, 8B aligned
  unsigned* maxbits = (unsigned*)(accum + 1);

  ssim_init_kernel<<<1, 32, 0, stream>>>(maxbits, accum);
  ssim_amp_kernel<<<AMPN / 256, 256, 0, stream>>>(Xt, Yt, mean, stdv,
                                                  Xa, Ya, maxbits);
  ssim_tile_kernel<<<NIMG * 256, 32, 0, stream>>>(Xa, Ya, maxbits, accum);
  ssim_final_kernel<<<1, 1, 0, stream>>>(accum, out);
}